// GineBlock_19868518711757
// MI455X (gfx1250) — compile-verified
//
#include <hip/hip_runtime.h>

typedef __attribute__((ext_vector_type(2))) float v2f;
typedef __attribute__((ext_vector_type(8))) float v8f;

#define N_NODES 100000
#define N_EDGES 1600000
#define D 128
#define BN_EPS 1e-5f

// ---------------------------------------------------------------------------
// Zero agg + sums + sumsq (contiguous first region of workspace)
// ---------------------------------------------------------------------------
__global__ void zero_kernel(float* __restrict__ p, long long n4) {
    long long i = (long long)blockIdx.x * blockDim.x + threadIdx.x;
    if (i < n4) {
        ((float4*)p)[i] = make_float4(0.f, 0.f, 0.f, 0.f);
    }
}

// ---------------------------------------------------------------------------
// Edge phase: one wave per edge, 4 floats per lane.
// msg = relu(x[src] + e);  atomicAdd into agg[dst]
// ---------------------------------------------------------------------------
__global__ __launch_bounds__(256)
void edge_scatter(const float* __restrict__ x,
                  const long long* __restrict__ ei,
                  const float* __restrict__ ea,
                  float* __restrict__ agg) {
    const int e    = blockIdx.x * 8 + (threadIdx.x >> 5);   // 8 edges per block
    const int lane = threadIdx.x & 31;
    const int c    = lane * 4;

    const int s = (int)ei[e];
    const int d = (int)ei[N_EDGES + e];

    const float4 xv = *(const float4*)(x  + (long long)s * D + c);
    const float4 ev = *(const float4*)(ea + (long long)e * D + c);

    float4 m;
    m.x = fmaxf(xv.x + ev.x, 0.f);
    m.y = fmaxf(xv.y + ev.y, 0.f);
    m.z = fmaxf(xv.z + ev.z, 0.f);
    m.w = fmaxf(xv.w + ev.w, 0.f);

    float* dp = agg + (long long)d * D + c;
    atomicAdd(dp + 0, m.x);
    atomicAdd(dp + 1, m.y);
    atomicAdd(dp + 2, m.z);
    atomicAdd(dp + 3, m.w);
}

// ---------------------------------------------------------------------------
// Fused MLP: h2 = relu((x+agg) @ W1 + b1) @ W2 + b2, with per-feature
// sum / sumsq reduction for BatchNorm.
// Block = 256 threads = 8 waves. Wave w owns output columns [16w, 16w+16).
// W1/W2 fragments stay resident in VGPRs across the grid-stride row loop.
// fp32 WMMA 16x16x4 keeps full reference precision.
// ---------------------------------------------------------------------------
__global__ __launch_bounds__(256)
void mlp_kernel(const float* __restrict__ x,
                const float* __restrict__ agg,
                const float* __restrict__ W1, const float* __restrict__ b1,
                const float* __restrict__ W2, const float* __restrict__ b2,
                float* __restrict__ h2,
                float* __restrict__ sums, float* __restrict__ sumsq) {
    __shared__ float hs[16][132];   // x+agg tile  (pad 132 -> conflict-free)
    __shared__ float ts[16][132];   // relu(GEMM1) tile

    const int tid  = threadIdx.x;
    const int w    = tid >> 5;          // wave id 0..7
    const int lane = tid & 31;
    const int half = lane >> 4;         // 0: lanes 0-15, 1: lanes 16-31
    const int l16  = lane & 15;
    const int n0   = w * 16;
    const int col  = n0 + l16;          // output column owned by this lane

    // --- load resident B fragments (ISA 16x4 A-layout mirrored for 4x16 B) ---
    v2f bw1[32], bw2[32];
#pragma unroll
    for (int kk = 0; kk < 32; ++kk) {
        const int kb = kk * 4 + half * 2;
        bw1[kk] = (v2f){ W1[kb * D + col], W1[(kb + 1) * D + col] };
        bw2[kk] = (v2f){ W2[kb * D + col], W2[(kb + 1) * D + col] };
    }
    const float b1c = b1[col];
    const float b2c = b2[col];

    float gsum = 0.f, gsq = 0.f;

    for (int tile = blockIdx.x; tile < N_NODES / 16; tile += gridDim.x) {
        const int row0 = tile * 16;

        // stage h = x + agg into LDS (each thread: 8 contiguous floats)
        {
            const int r  = tid >> 4;
            const int c0 = (tid & 15) * 8;
            const float4* xp = (const float4*)(x   + (long long)(row0 + r) * D + c0);
            const float4* ap = (const float4*)(agg + (long long)(row0 + r) * D + c0);
            float4 a0 = xp[0], a1 = xp[1];
            float4 g0 = ap[0], g1 = ap[1];
            hs[r][c0 + 0] = a0.x + g0.x;  hs[r][c0 + 1] = a0.y + g0.y;
            hs[r][c0 + 2] = a0.z + g0.z;  hs[r][c0 + 3] = a0.w + g0.w;
            hs[r][c0 + 4] = a1.x + g1.x;  hs[r][c0 + 5] = a1.y + g1.y;
            hs[r][c0 + 6] = a1.z + g1.z;  hs[r][c0 + 7] = a1.w + g1.w;
        }
        __syncthreads();

        // --- GEMM1: t = relu(h @ W1 + b1) ---
        v8f acc = {};
#pragma unroll
        for (int kk = 0; kk < 32; ++kk) {
            const int kbA = kk * 4 + half * 2;
            v2f a = (v2f){ hs[l16][kbA], hs[l16][kbA + 1] };
            acc = __builtin_amdgcn_wmma_f32_16x16x4_f32(
                false, a, false, bw1[kk], (short)0, acc, false, false);
        }
#pragma unroll
        for (int v = 0; v < 8; ++v) {
            // D layout: VGPR v -> row M = v + 8*half, col = n0 + l16
            ts[v + half * 8][col] = fmaxf(acc[v] + b1c, 0.f);
        }
        __syncthreads();

        // --- GEMM2: h2 = t @ W2 + b2 ---
        v8f acc2 = {};
#pragma unroll
        for (int kk = 0; kk < 32; ++kk) {
            const int kbA = kk * 4 + half * 2;
            v2f a = (v2f){ ts[l16][kbA], ts[l16][kbA + 1] };
            acc2 = __builtin_amdgcn_wmma_f32_16x16x4_f32(
                false, a, false, bw2[kk], (short)0, acc2, false, false);
        }
#pragma unroll
        for (int v = 0; v < 8; ++v) {
            const float hv = acc2[v] + b2c;
            h2[(long long)(row0 + v + half * 8) * D + col] = hv;
            gsum += hv;
            gsq  += hv * hv;
        }
        // next iteration's hs write is safe: everyone passed the ts barrier,
        // and hs writes precede the top-of-loop barrier.
    }

    // per-lane partial stats all belong to a single column
    atomicAdd(&sums[col],  gsum);
    atomicAdd(&sumsq[col], gsq);
}

// ---------------------------------------------------------------------------
// BatchNorm (training stats, biased var) + ReLU
// ---------------------------------------------------------------------------
__global__ __launch_bounds__(256)
void bn_relu(const float* __restrict__ h2,
             const float* __restrict__ sums, const float* __restrict__ sumsq,
             const float* __restrict__ gamma, const float* __restrict__ beta,
             float* __restrict__ out) {
    const long long i = ((long long)blockIdx.x * blockDim.x + threadIdx.x) * 4;
    if (i >= (long long)N_NODES * D) return;
    const int c = (int)(i & (D - 1));
    const float invN = 1.0f / (float)N_NODES;

    float4 v = *(const float4*)(h2 + i);
    float r[4] = { v.x, v.y, v.z, v.w };
#pragma unroll
    for (int j = 0; j < 4; ++j) {
        const int cc = c + j;
        const float mean = sums[cc] * invN;
        const float var  = sumsq[cc] * invN - mean * mean;
        const float inv  = rsqrtf(var + BN_EPS);
        r[j] = fmaxf((r[j] - mean) * inv * gamma[cc] + beta[cc], 0.f);
    }
    *(float4*)(out + i) = make_float4(r[0], r[1], r[2], r[3]);
}

// ---------------------------------------------------------------------------
extern "C" void kernel_launch(void* const* d_in, const int* in_sizes, int n_in,
                              void* d_out, int out_size, void* d_ws, size_t ws_size,
                              hipStream_t stream) {
    const float*     x     = (const float*)d_in[0];
    const long long* ei    = (const long long*)d_in[1];
    const float*     ea    = (const float*)d_in[2];
    const float*     W1    = (const float*)d_in[3];
    const float*     b1    = (const float*)d_in[4];
    const float*     W2    = (const float*)d_in[5];
    const float*     b2    = (const float*)d_in[6];
    const float*     gamma = (const float*)d_in[7];
    const float*     beta  = (const float*)d_in[8];
    float*           out   = (float*)d_out;

    // workspace layout: agg | sums | sumsq | h2   (agg..sumsq zeroed each call)
    float* agg   = (float*)d_ws;
    float* sums  = agg + (size_t)N_NODES * D;
    float* sumsq = sums + D;
    float* h2    = sumsq + D;

    const long long zero_elems = (long long)N_NODES * D + 2 * D;   // 12,800,256
    const long long zero4      = zero_elems / 4;                   // 3,200,064
    zero_kernel<<<(int)((zero4 + 255) / 256), 256, 0, stream>>>(agg, zero4);

    edge_scatter<<<N_EDGES / 8, 256, 0, stream>>>(x, ei, ea, agg);

    mlp_kernel<<<512, 256, 0, stream>>>(x, agg, W1, b1, W2, b2, h2, sums, sumsq);

    bn_relu<<<(N_NODES * D / 4 + 255) / 256, 256, 0, stream>>>(
        h2, sums, sumsq, gamma, beta, out);
}